// weighted_graph_conv_38465727103769
// MI455X (gfx1250) — compile-verified
//
#include <hip/hip_runtime.h>

typedef __attribute__((ext_vector_type(2))) float v2f;
typedef __attribute__((ext_vector_type(8))) float v8f;

#define T_STEPS 4
#define F_DIM   64

// ---------------------------------------------------------------------------
// Phase 0: zero the h accumulator (we reuse d_out as h storage; it is fully
// overwritten by phase 2, and d_out/d_ws arrive poisoned).
// ---------------------------------------------------------------------------
__global__ void wgc_zero_kernel(float4* __restrict__ h4, int n4) {
  int i = blockIdx.x * blockDim.x + threadIdx.x;
  int stride = gridDim.x * blockDim.x;
  for (; i < n4; i += stride) {
    h4[i] = make_float4(0.f, 0.f, 0.f, 0.f);
  }
}

// ---------------------------------------------------------------------------
// Phase 1: msg[e,t,f] = ew[t,e] * nf[src[e],t,f]; scatter-add into h[dst[e],t,f].
// One 256-thread block per edge: tid -> (t = tid>>6, f = tid&63).
// Gather of nf row is coalesced (64 consecutive floats per (t) group);
// atomic adds land in L2 (h = 51MB, fits in 192MB L2).
// ---------------------------------------------------------------------------
__global__ void wgc_scatter_kernel(const float* __restrict__ nf,
                                   const float* __restrict__ ew,
                                   const int* __restrict__ src,
                                   const int* __restrict__ dst,
                                   float* __restrict__ h,
                                   int E) {
  int e = blockIdx.x;
  int tid = threadIdx.x;
  int t = tid >> 6;
  int f = tid & 63;
  int s = src[e];
  int d = dst[e];
  float w = ew[t * E + e];
  float x = nf[(s * T_STEPS + t) * F_DIM + f];
  atomicAdd(&h[(d * T_STEPS + t) * F_DIM + f], w * x);
}

// ---------------------------------------------------------------------------
// Phase 2: Out = H (rows x 64) * W^T (64 x 64) + bias, in f32 via
// V_WMMA_F32_16X16X4_F32. One wave handles a 16-row tile in place:
// full A-tile preloaded to registers before any store -> in-place safe.
//
// Layouts (ISA 7.12.2, 16x16x4 f32):
//   A (16x4):  lane m = L&15 holds row m; VGPR0 = K {0|2}, VGPR1 = K {1|3}
//              selected by half = L>>4  -> v2f at H[m, k0 + 2*half]
//   B (4x16):  lane n = L&15 holds col n; same K striping -> v2f from
//              weight[o = ntile*16 + n][k0 + 2*half]   (B[k][n] = W^T[k][o] = weight[o][k])
//   C/D 16x16: VGPR v, lane L -> element [row = v + 8*half][col = L&15]
// ---------------------------------------------------------------------------
__global__ __launch_bounds__(256) void wgc_gemm_kernel(float* __restrict__ h_out,
                                                       const float* __restrict__ weight,
                                                       const float* __restrict__ bias,
                                                       int num_tiles) {
  int wave = blockIdx.x * (blockDim.x >> 5) + (threadIdx.x >> 5);
  int lane = threadIdx.x & 31;
  if (wave >= num_tiles) return;  // wave-uniform: EXEC stays all-ones for WMMA

  int m = lane & 15;
  int half = lane >> 4;
  int rowbase = wave * 16;

  // Preload the full 16x64 A tile (16 k-steps of v2f per lane = whole tile in wave)
  const float* hrow = h_out + (rowbase + m) * F_DIM;
  v2f A[16];
#pragma unroll
  for (int k = 0; k < 16; ++k) {
    A[k] = *(const v2f*)(hrow + k * 4 + 2 * half);
  }

  const float* wrow = weight;  // weight[o][f], o-major, 64 floats per row
  float bcol_base;             // bias for this lane's output column, per ntile

#pragma unroll
  for (int nt = 0; nt < 4; ++nt) {
    int ocol = nt * 16 + (lane & 15);
    bcol_base = bias[ocol];
    v8f acc = {bcol_base, bcol_base, bcol_base, bcol_base,
               bcol_base, bcol_base, bcol_base, bcol_base};

    const float* wr = wrow + ocol * F_DIM;
#pragma unroll
    for (int k = 0; k < 16; ++k) {
      v2f B = *(const v2f*)(wr + k * 4 + 2 * half);
      // 8 args: (neg_a, A, neg_b, B, c_mod, C, reuse_a, reuse_b)
      acc = __builtin_amdgcn_wmma_f32_16x16x4_f32(
          /*neg_a=*/false, A[k], /*neg_b=*/false, B,
          /*c_mod=*/(short)0, acc, /*reuse_a=*/false, /*reuse_b=*/false);
    }

    // Store D: VGPR v, lane -> [rowbase + v + 8*half][nt*16 + (lane&15)]
#pragma unroll
    for (int v = 0; v < 8; ++v) {
      int mm = v + 8 * half;
      h_out[(rowbase + mm) * F_DIM + ocol] = acc[v];
    }
  }
}

// ---------------------------------------------------------------------------
extern "C" void kernel_launch(void* const* d_in, const int* in_sizes, int n_in,
                              void* d_out, int out_size, void* d_ws, size_t ws_size,
                              hipStream_t stream) {
  const float* nf     = (const float*)d_in[0];  // (N, T, 64)
  const float* ew     = (const float*)d_in[1];  // (T, E)
  const int*   src    = (const int*)d_in[2];    // (E,)
  const int*   dst    = (const int*)d_in[3];    // (E,)
  const float* weight = (const float*)d_in[4];  // (64, 64)
  const float* bias   = (const float*)d_in[5];  // (64,)
  float* out = (float*)d_out;                   // (N, T, 64), doubles as h accumulator

  int E = in_sizes[2];
  int N = in_sizes[0] / (T_STEPS * F_DIM);
  int rows = N * T_STEPS;          // 200000
  int h_elems = rows * F_DIM;      // 12.8M floats

  // Phase 0: zero h (== d_out)
  {
    int n4 = h_elems / 4;
    int blocks = 2048;
    wgc_zero_kernel<<<blocks, 256, 0, stream>>>((float4*)out, n4);
  }

  // Phase 1: gather * edge-weight, atomic scatter-sum
  wgc_scatter_kernel<<<E, 256, 0, stream>>>(nf, ew, src, dst, out, E);

  // Phase 2: in-place f32 WMMA GEMM + bias
  {
    int num_tiles = rows / 16;     // 12500 (exact for N=50000)
    int waves_per_block = 8;       // 256 threads
    int blocks = (num_tiles + waves_per_block - 1) / waves_per_block;
    wgc_gemm_kernel<<<blocks, 256, 0, stream>>>(out, weight, bias, num_tiles);
  }
}